// SparseKernelConv_8065948582435
// MI455X (gfx1250) — compile-verified
//
#include <hip/hip_runtime.h>

typedef __attribute__((ext_vector_type(2))) float v2f;
typedef __attribute__((ext_vector_type(8))) float v8f;

#define TM 64          // output rows per workgroup tile
#define NK 27          // kernel offsets
#define CIN 64
#define COUT 64
#define LDS_STRIDE 68  // 64 + 4 pad -> conflict-free ds_load_b64 A reads
#define BUFSZ (TM * LDS_STRIDE)

// ---- one-time weight repack: R[k][cc][half][n] = {W[k][4cc+2h][n], W[k][4cc+2h+1][n]} ----
__global__ __launch_bounds__(256)
void repack_weights(const float* __restrict__ W, float* __restrict__ R) {
    int p = blockIdx.x * 256 + threadIdx.x;          // pair index
    if (p >= NK * 16 * 2 * COUT) return;
    int n  = p & 63;
    int h  = (p >> 6) & 1;
    int cc = (p >> 7) & 15;
    int k  = p >> 11;
    int row = cc * 4 + 2 * h;
    float2 v;
    v.x = W[(k * CIN + row) * COUT + n];
    v.y = W[(k * CIN + row + 1) * COUT + n];
    ((float2*)R)[p] = v;
}

template <bool PACKED>
__global__ __launch_bounds__(128)
void spconv_wmma_f32(const float* __restrict__ feats,
                     const long long* __restrict__ rulebook,
                     const float* __restrict__ weight,   // (27, 64, 64)
                     const float* __restrict__ wrep,     // repacked pairs (or null)
                     const float* __restrict__ bias,     // (64,)
                     float* __restrict__ out,            // (M, 64)
                     int M) {
    __shared__ float sm_g[2 * BUFSZ];         // double-buffered gathered 64x64 tile
    __shared__ int   sm_idx[TM * NK];         // rulebook slice (narrowed)
    __shared__ float sm_inv[TM];              // 1/denom per row

    const int tid  = threadIdx.x;
    const int lane = tid & 31;
    const int wave = tid >> 5;
    const int gm   = blockIdx.x * TM;

    // ---- load the 64x27 rulebook slice into LDS (int64 -> int32) ----
    for (int t = tid; t < TM * NK; t += 128) {
        int r  = t / NK;
        int kk = t - r * NK;
        int rr = gm + r; if (rr > M - 1) rr = M - 1;
        sm_idx[t] = (int)rulebook[(long long)rr * NK + kk];
    }
    __syncthreads();

    // ---- per-row valid-neighbor count -> 1/denom ----
    if (tid < TM) {
        int cnt = 0;
        #pragma unroll
        for (int kk = 0; kk < NK; ++kk) cnt += (sm_idx[tid * NK + kk] >= 0) ? 1 : 0;
        sm_inv[tid] = 1.0f / (float)(cnt < 1 ? 1 : cnt);
    }

    // staging map: thread -> (row 0..63, 32-float half-row segment 0..1)
    const int srow = tid >> 1;
    const int scol = (tid & 1) * 32;

    float4 g[8];
    #pragma unroll
    for (int j = 0; j < 8; ++j) g[j] = make_float4(0.f, 0.f, 0.f, 0.f);
    {
        int idx = sm_idx[srow * NK + 0];
        if (idx >= 0) {
            const float4* src = (const float4*)(feats + (long long)idx * CIN + scol);
            #pragma unroll
            for (int j = 0; j < 8; ++j) g[j] = src[j];
        }
    }
    // commit k=0 tile to buffer 0
    {
        float4* dst = (float4*)(&sm_g[srow * LDS_STRIDE + scol]);
        #pragma unroll
        for (int j = 0; j < 8; ++j) dst[j] = g[j];
    }
    __syncthreads();

    // wave -> 2 row-groups x 2 cout-tiles (4 accumulators)
    const int rg   = (wave >> 1) * 32;   // rows rg..rg+31
    const int ct   = (wave & 1) * 32;    // couts ct..ct+31
    const int half = lane >> 4;
    const int ml   = lane & 15;

    v8f acc00 = {}, acc01 = {}, acc10 = {}, acc11 = {};

    for (int k = 0; k < NK; ++k) {
        const float* buf = &sm_g[(k & 1) * BUFSZ];

        // ---- issue gather loads for k+1 (overlap with WMMA below) ----
        #pragma unroll
        for (int j = 0; j < 8; ++j) g[j] = make_float4(0.f, 0.f, 0.f, 0.f);
        if (k + 1 < NK) {
            int idx = sm_idx[srow * NK + k + 1];
            if (idx >= 0) {
                const float4* src = (const float4*)(feats + (long long)idx * CIN + scol);
                #pragma unroll
                for (int j = 0; j < 8; ++j) g[j] = src[j];
            }
        }

        const float*  wk  = weight + k * (CIN * COUT);
        const float2* wkp = (const float2*)wrep + (k * 16) * 2 * COUT;

        // ---- 16 cin-chunks of 4: 2 A frags x 2 B frags -> 4 WMMAs ----
        #pragma unroll
        for (int cc = 0; cc < CIN / 4; ++cc) {
            const int c0 = cc * 4 + 2 * half;
            v2f a0 = *(const v2f*)(&buf[(rg + ml) * LDS_STRIDE + c0]);
            v2f a1 = *(const v2f*)(&buf[(rg + 16 + ml) * LDS_STRIDE + c0]);
            v2f b0, b1;
            if (PACKED) {
                const float2* bp = wkp + (cc * 2 + half) * COUT;
                float2 p0 = bp[ct + ml];
                float2 p1 = bp[ct + 16 + ml];
                b0.x = p0.x; b0.y = p0.y;
                b1.x = p1.x; b1.y = p1.y;
            } else {
                b0.x = wk[(c0 + 0) * COUT + ct + ml];
                b0.y = wk[(c0 + 1) * COUT + ct + ml];
                b1.x = wk[(c0 + 0) * COUT + ct + 16 + ml];
                b1.y = wk[(c0 + 1) * COUT + ct + 16 + ml];
            }
            acc00 = __builtin_amdgcn_wmma_f32_16x16x4_f32(
                false, a0, false, b0, (short)0, acc00, false, false);
            acc01 = __builtin_amdgcn_wmma_f32_16x16x4_f32(
                false, a0, false, b1, (short)0, acc01, false, false);
            acc10 = __builtin_amdgcn_wmma_f32_16x16x4_f32(
                false, a1, false, b0, (short)0, acc10, false, false);
            acc11 = __builtin_amdgcn_wmma_f32_16x16x4_f32(
                false, a1, false, b1, (short)0, acc11, false, false);
        }

        // ---- commit k+1 tile into the other buffer ----
        if (k + 1 < NK) {
            float4* dst = (float4*)(&sm_g[((k + 1) & 1) * BUFSZ + srow * LDS_STRIDE + scol]);
            #pragma unroll
            for (int j = 0; j < 8; ++j) dst[j] = g[j];
        }
        __syncthreads();   // single barrier per k (double-buffered)
    }

    // ---- epilogue: /denom + bias + residual, per documented C/D layout ----
    const int col0 = ct + ml;
    const int col1 = ct + 16 + ml;
    #pragma unroll
    for (int i = 0; i < 8; ++i) {
        int rl0 = rg + i + 8 * half;       // acc0x rows
        int rl1 = rl0 + 16;                // acc1x rows
        long long grow0 = (long long)gm + rl0;
        long long grow1 = (long long)gm + rl1;
        if (grow0 < M) {
            long long o0 = grow0 * COUT + col0;
            long long o1 = grow0 * COUT + col1;
            out[o0] = acc00[i] * sm_inv[rl0] + bias[col0] + feats[o0];
            out[o1] = acc01[i] * sm_inv[rl0] + bias[col1] + feats[o1];
        }
        if (grow1 < M) {
            long long o0 = grow1 * COUT + col0;
            long long o1 = grow1 * COUT + col1;
            out[o0] = acc10[i] * sm_inv[rl1] + bias[col0] + feats[o0];
            out[o1] = acc11[i] * sm_inv[rl1] + bias[col1] + feats[o1];
        }
    }
}

extern "C" void kernel_launch(void* const* d_in, const int* in_sizes, int n_in,
                              void* d_out, int out_size, void* d_ws, size_t ws_size,
                              hipStream_t stream) {
    const float*     feats    = (const float*)d_in[0];
    const long long* rulebook = (const long long*)d_in[1];
    const float*     weight   = (const float*)d_in[2];
    const float*     bias     = (const float*)d_in[3];
    float*           out      = (float*)d_out;

    const int M = in_sizes[0] / CIN;   // 100000
    const int grid = (M + TM - 1) / TM;

    const size_t need = (size_t)NK * CIN * COUT * sizeof(float);  // 442 KB
    if (ws_size >= need) {
        float* wrep = (float*)d_ws;
        int pairs = NK * 16 * 2 * COUT;
        repack_weights<<<(pairs + 255) / 256, 256, 0, stream>>>(weight, wrep);
        spconv_wmma_f32<true><<<grid, 128, 0, stream>>>(
            feats, rulebook, weight, wrep, bias, out, M);
    } else {
        spconv_wmma_f32<false><<<grid, 128, 0, stream>>>(
            feats, rulebook, weight, nullptr, bias, out, M);
    }
}